// point_rnn_63196148793619
// MI455X (gfx1250) — compile-verified
//
#include <hip/hip_runtime.h>

typedef __attribute__((ext_vector_type(2))) float v2f;
typedef __attribute__((ext_vector_type(8))) float v8f;

#define BATCH   2
#define NPTS    8192
#define FEAT    128
#define OUTC    128
#define K_NN    16
#define CIN     259
#define KRED    132          // reduced K: 128 (S2 or X1) + 3 (P2 or -P1) + 1 pad
#define KSTEPS  33           // KRED / 4
#define QPB     16           // queries per block in main kernel

// workspace layout (bytes)
#define WS_KNN_OFF  0                                   // int[B*N*16]      = 1,048,576 B
#define WS_WT1_OFF  (BATCH*NPTS*K_NN*4)                 // float[132*128]   =    67,584 B
#define WS_WT2_OFF  (WS_WT1_OFF + KRED*OUTC*4)
#define WS_CST_OFF  (WS_WT2_OFF + KRED*OUTC*4)          // float[B*N*128]   = 8,388,608 B
// total ~9.2 MB

// ---------------------------------------------------------------------------
// Kernel 0: repack W (OUT x 259, row-major) into K-major operand arrays.
//   Wt1[k][o] = k<128 ? W[o][k]      : k<131 ? W[o][256+k-128] : 0   (S2 | +P2 path)
//   Wt2[k][o] = k<128 ? W[o][128+k]  : k<131 ? W[o][256+k-128] : 0   (X1 | -P1 path)
// ---------------------------------------------------------------------------
__global__ void prep_w(const float* __restrict__ W,
                       float* __restrict__ wt1, float* __restrict__ wt2) {
    int t = blockIdx.x * 256 + threadIdx.x;
    if (t >= KRED * OUTC) return;
    int k = t / OUTC, o = t % OUTC;
    float v1 = 0.f, v2 = 0.f;
    if (k < 128)      { v1 = W[o * CIN + k];            v2 = W[o * CIN + 128 + k]; }
    else if (k < 131) { v1 = v2 = W[o * CIN + 256 + (k - 128)]; }
    wt1[k * OUTC + o] = v1;
    wt2[k * OUTC + o] = v2;
}

// ---------------------------------------------------------------------------
// Kernel 1: KNN. One thread per query point; P2 tiles staged in LDS.
// Register-resident sorted top-16 (ascending d2) via guarded bubble insert.
// ---------------------------------------------------------------------------
#define KNN_TILE 512
__global__ void knn_kernel(const float* __restrict__ P1,
                           const float* __restrict__ P2,
                           int* __restrict__ knn) {
    __shared__ __align__(16) float4 tile[KNN_TILE];
    const int b = blockIdx.y;
    const int n = blockIdx.x * blockDim.x + threadIdx.x;
    const float* p1 = P1 + ((size_t)b * NPTS + n) * 3;
    const float qx = p1[0], qy = p1[1], qz = p1[2];
    const float* base = P2 + (size_t)b * NPTS * 3;

    float bd[K_NN]; int bi[K_NN];
#pragma unroll
    for (int j = 0; j < K_NN; ++j) { bd[j] = 3.4e38f; bi[j] = 0; }

    for (int t0 = 0; t0 < NPTS; t0 += KNN_TILE) {
        __syncthreads();
        for (int c = threadIdx.x; c < KNN_TILE; c += blockDim.x) {
            int j = t0 + c;
            tile[c] = make_float4(base[3*j], base[3*j+1], base[3*j+2], 0.f);
        }
        __syncthreads();
        for (int c = 0; c < KNN_TILE; ++c) {
            float4 p = tile[c];
            float dx = p.x - qx, dy = p.y - qy, dz = p.z - qz;
            float d2 = dx*dx + dy*dy + dz*dz;
            if (d2 < bd[K_NN - 1]) {          // rare after warm-up
                int id = t0 + c;
#pragma unroll
                for (int j = 0; j < K_NN; ++j) {
                    if (d2 < bd[j]) {
                        float td = bd[j]; bd[j] = d2; d2 = td;
                        int   ti = bi[j]; bi[j] = id; id = ti;
                    }
                }
            }
        }
    }
    int* outp = knn + ((size_t)b * NPTS + n) * K_NN;
#pragma unroll
    for (int j = 0; j < K_NN; ++j) outp[j] = bi[j];
}

// ---------------------------------------------------------------------------
// Kernel 2: per-query constant vector via WMMA:
//   Cst[q][o] = X1[q]·W[o][128:256] - P1[q]·W[o][256:259] + bias[o]
// Block: 256 threads = 8 waves = 8 N-tiles; 16 queries per block (M dim).
// ---------------------------------------------------------------------------
__global__ void const_kernel(const float* __restrict__ P1,
                             const float* __restrict__ X1,
                             const float* __restrict__ bias,
                             const float* __restrict__ wt2,
                             float* __restrict__ cst) {
    __shared__ __align__(16) float Abuf[16 * KRED];
    const int q0   = blockIdx.x * 16;
    const int tid  = threadIdx.x;
    const int lane = tid & 31;
    const int wave = tid >> 5;
    const int N0   = wave * 16;
    const int col  = lane & 15;
    const int krow = (lane >> 4) * 2;

    {   // cooperative stage of A panel: 16 rows (queries) x 132
        int r = tid >> 4, k = tid & 15;
        int q = q0 + r;
        const float* x1row = X1 + (size_t)q * FEAT;
        *reinterpret_cast<float4*>(&Abuf[r*KRED +      4*k]) = *reinterpret_cast<const float4*>(x1row +      4*k);
        *reinterpret_cast<float4*>(&Abuf[r*KRED + 64 + 4*k]) = *reinterpret_cast<const float4*>(x1row + 64 + 4*k);
        if (k == 0) {
            const float* p1 = P1 + (size_t)q * 3;
            Abuf[r*KRED + 128] = -p1[0];
            Abuf[r*KRED + 129] = -p1[1];
            Abuf[r*KRED + 130] = -p1[2];
            Abuf[r*KRED + 131] = 0.f;
        }
    }

    // B fragments (4x16 f32 layout: VGPR0 -> K=krow, VGPR1 -> K=krow+1)
    v2f Bfrag[KSTEPS];
#pragma unroll
    for (int s = 0; s < KSTEPS; ++s) {
        int k0 = 4*s + krow;
        Bfrag[s].x = wt2[(k0    ) * OUTC + N0 + col];
        Bfrag[s].y = wt2[(k0 + 1) * OUTC + N0 + col];
    }
    __syncthreads();

    v8f acc = {0.f,0.f,0.f,0.f,0.f,0.f,0.f,0.f};
#pragma unroll
    for (int s = 0; s < KSTEPS; ++s) {
        const float2 af = *reinterpret_cast<const float2*>(&Abuf[col*KRED + 4*s + krow]);
        v2f a; a.x = af.x; a.y = af.y;
        acc = __builtin_amdgcn_wmma_f32_16x16x4_f32(false, a, false, Bfrag[s],
                                                    (short)0, acc, false, false);
    }

    float bb = bias[N0 + col];
#pragma unroll
    for (int v = 0; v < 8; ++v) {
        int m = v + ((lane >> 4) << 3);   // lanes 0-15: M=v ; lanes 16-31: M=v+8
        cst[(size_t)(q0 + m) * OUTC + N0 + col] = acc[v] + bb;
    }
}

// ---------------------------------------------------------------------------
// Kernel 3: fused gather + 16x132 @ 132x128 WMMA GEMM + max over 16 neighbors.
// Block: 256 threads = 8 waves (one 16-wide N-tile each), QPB queries per block.
// B fragments (Wt1) persistent in registers across all queries of the block.
// ---------------------------------------------------------------------------
__global__ void main_kernel(const float* __restrict__ P2,
                            const float* __restrict__ S2,
                            const int*   __restrict__ knn,
                            const float* __restrict__ wt1,
                            const float* __restrict__ cst,
                            float* __restrict__ out) {
    __shared__ __align__(16) float Abuf[16 * KRED];
    const int tid  = threadIdx.x;
    const int lane = tid & 31;
    const int wave = tid >> 5;
    const int N0   = wave * 16;
    const int col  = lane & 15;
    const int krow = (lane >> 4) * 2;

    v2f Bfrag[KSTEPS];
#pragma unroll
    for (int s = 0; s < KSTEPS; ++s) {
        int k0 = 4*s + krow;
        Bfrag[s].x = wt1[(k0    ) * OUTC + N0 + col];
        Bfrag[s].y = wt1[(k0 + 1) * OUTC + N0 + col];
    }

    const int r  = tid >> 4;   // neighbor row this thread helps stage
    const int kk = tid & 15;

    for (int qi = 0; qi < QPB; ++qi) {
        const int q = blockIdx.x * QPB + qi;
        const int b = q >> 13;                    // q / NPTS
        __syncthreads();
        {   // stage A panel: rows = 16 gathered neighbors, cols = [S2(128) | P2(3) | 0]
            int j = knn[(size_t)q * K_NN + r];
            const float* s2row = S2 + ((size_t)b * NPTS + j) * FEAT;
            *reinterpret_cast<float4*>(&Abuf[r*KRED +      4*kk]) = *reinterpret_cast<const float4*>(s2row +      4*kk);
            *reinterpret_cast<float4*>(&Abuf[r*KRED + 64 + 4*kk]) = *reinterpret_cast<const float4*>(s2row + 64 + 4*kk);
            if (kk == 0) {
                const float* p2 = P2 + ((size_t)b * NPTS + j) * 3;
                Abuf[r*KRED + 128] = p2[0];
                Abuf[r*KRED + 129] = p2[1];
                Abuf[r*KRED + 130] = p2[2];
                Abuf[r*KRED + 131] = 0.f;
            }
        }
        __syncthreads();

        v8f acc = {0.f,0.f,0.f,0.f,0.f,0.f,0.f,0.f};
#pragma unroll
        for (int s = 0; s < KSTEPS; ++s) {
            const float2 af = *reinterpret_cast<const float2*>(&Abuf[col*KRED + 4*s + krow]);
            v2f a; a.x = af.x; a.y = af.y;
            acc = __builtin_amdgcn_wmma_f32_16x16x4_f32(false, a, false, Bfrag[s],
                                                        (short)0, acc, false, false);
        }

        // max over M (16 neighbors): 8 per-lane rows, then combine lane L with L^16
        float m = acc[0];
#pragma unroll
        for (int v = 1; v < 8; ++v) m = fmaxf(m, acc[v]);
        m = fmaxf(m, __shfl_xor(m, 16, 32));
        m += cst[(size_t)q * OUTC + N0 + col];
        if (lane < 16) out[(size_t)q * OUTC + N0 + lane] = m;
    }
}

// ---------------------------------------------------------------------------
extern "C" void kernel_launch(void* const* d_in, const int* in_sizes, int n_in,
                              void* d_out, int out_size, void* d_ws, size_t ws_size,
                              hipStream_t stream) {
    const float* P1   = (const float*)d_in[0];
    const float* P2   = (const float*)d_in[1];
    const float* X1   = (const float*)d_in[2];
    const float* S2   = (const float*)d_in[3];
    const float* W    = (const float*)d_in[4];
    const float* bias = (const float*)d_in[5];
    float* outp = (float*)d_out;

    char*  ws  = (char*)d_ws;
    int*   knn = (int*)  (ws + WS_KNN_OFF);
    float* wt1 = (float*)(ws + WS_WT1_OFF);
    float* wt2 = (float*)(ws + WS_WT2_OFF);
    float* cst = (float*)(ws + WS_CST_OFF);

    prep_w     <<<(KRED*OUTC + 255)/256, 256, 0, stream>>>(W, wt1, wt2);
    knn_kernel <<<dim3(NPTS/256, BATCH), 256, 0, stream>>>(P1, P2, knn);
    const_kernel<<<(BATCH*NPTS)/16,      256, 0, stream>>>(P1, X1, bias, wt2, cst);
    main_kernel <<<(BATCH*NPTS)/QPB,     256, 0, stream>>>(P2, S2, knn, wt1, cst, outp);
}